// LocallyConnectedXYZLayer_10179072491794
// MI455X (gfx1250) — compile-verified
//
#include <hip/hip_runtime.h>
#include <stdint.h>

// Problem constants (from reference): N=4, C=20, H=64, W=2048, 5x5 kernel, sigma=1
#define NN 4
#define CC 20
#define HH 64
#define WW 2048
#define TILE 256
#define LDSW 264                 // staged row stride in elements (66 quads * 4)

// exp(-d2 / (2*sigma^2)) = exp2(-d2 * log2(e)/2)
#define KEXP 0.7213475204444817f

__device__ __forceinline__ uint32_t lds_off(const void* p) {
    // Generic pointer to a __shared__ object: low 32 bits are the LDS byte offset.
    return (uint32_t)(uintptr_t)p;
}

__device__ __forceinline__ void async_ld_b128(uint32_t lds, int voff, const void* saddr) {
    asm volatile("global_load_async_to_lds_b128 %0, %1, %2"
                 :: "v"(lds), "v"(voff), "s"(saddr) : "memory");
}
__device__ __forceinline__ void async_ld_b32(uint32_t lds, int voff, const void* saddr) {
    asm volatile("global_load_async_to_lds_b32 %0, %1, %2"
                 :: "v"(lds), "v"(voff), "s"(saddr) : "memory");
}
__device__ __forceinline__ void wait_async0() {
    asm volatile("s_wait_asynccnt 0x0" ::: "memory");
}
__device__ __forceinline__ void wait_async2() {
    asm volatile("s_wait_asynccnt 0x2" ::: "memory");
}

__device__ __forceinline__ int clampi(int v, int lo, int hi) {
    return v < lo ? lo : (v > hi ? hi : v);
}

__global__ __launch_bounds__(TILE)
void bilateral5x5_kernel(const float* __restrict__ xyz,
                         const float* __restrict__ sm,
                         const unsigned char* __restrict__ mask,
                         float* __restrict__ out) {
    // LDS staging. Flat layout: quad-lane q writes bytes [q*16, q*16+16) which is
    // exactly row-major [unit][264] since 66 quads * 16B = 264 floats per row.
    __shared__ float sXYZ[4096];          // 15 rows (3ch x 5r) * 264 = 3960 (+pad)
    __shared__ float sSM[2][2048];        // 5 rows * 264 = 1320 per channel (+pad), dbl buf
    __shared__ unsigned char sMSK[2048];  // 5 rows * 264 bytes = 1320 (+pad)

    const int t   = threadIdx.x;
    const int n   = blockIdx.z;
    const int y   = blockIdx.y;
    const int x0  = blockIdx.x * TILE;
    const int x   = x0 + t;
    const int gx0 = x0 - 4;               // 16B-aligned left edge of staged tile

    const uint32_t xyzL = lds_off(sXYZ);
    const uint32_t mskL = lds_off(sMSK);
    const uint32_t smL0 = lds_off(sSM[0]);
    const uint32_t smL1 = lds_off(sSM[1]);

    // ---------------- stage xyz halo tile: 990 quad-lanes, 4 x B128 --------------
#pragma unroll
    for (int k = 0; k < 4; ++k) {
        int q  = t + k * 256;             // 0..1023 (LDS dest slot, unique)
        int cq = q > 989 ? 989 : q;       // clamp padding lanes onto valid data
        int unit = cq / 66;               // 0..14 = ch*5 + r
        int quad = cq - unit * 66;
        int ch = unit / 5;
        int r  = unit - ch * 5;
        int gy = clampi(y - 2 + r, 0, HH - 1);
        int gq = clampi(gx0 + quad * 4, 0, WW - 4);   // stays 16B aligned
        int voff = (((n * 3 + ch) * HH + gy) * WW + gq) * 4;
        async_ld_b128(xyzL + (uint32_t)q * 16u, voff, xyz);
    }

    // ---------------- stage mask bytes: 330 word-lanes, 2 x B32 -------------------
    int voff_plane[2]; // per-thread byte offsets inside one (H*W) fp32 plane
#pragma unroll
    for (int k = 0; k < 2; ++k) {
        int q  = t + k * 256;             // 0..511
        int cq = q > 329 ? 329 : q;
        int r    = cq / 66;
        int quad = cq - r * 66;
        int gy = clampi(y - 2 + r, 0, HH - 1);
        int gq = clampi(gx0 + quad * 4, 0, WW - 4);
        voff_plane[k] = (gy * WW + gq) * 4;           // fp32 plane offset (for sm)
        int moff = (n * HH + gy) * WW + gq;           // byte offset, 4B aligned
        async_ld_b32(mskL + (uint32_t)q * 4u, moff, mask);
    }

    // ---------------- stage softmax channel 0: 2 x B128 ---------------------------
    const float* smBase = sm + (size_t)(n * CC) * HH * WW;
#pragma unroll
    for (int k = 0; k < 2; ++k)
        async_ld_b128(smL0 + (uint32_t)(t + k * 256) * 16u, voff_plane[k], smBase);

    // xyz + mask (6 issues) done; channel-0 softmax (2) may remain in flight
    wait_async2();
    __syncthreads();

    // ---------------- per-pixel 5x5 weights (mask & bounds folded in) -------------
    // staged col c <-> global x0-4+c ; center x=x0+t <-> col t+4 ; tap dx -> col t+2+dx
    const int ccol = t + 4;
    float cx = sXYZ[(0 * 5 + 2) * LDSW + ccol];
    float cy = sXYZ[(1 * 5 + 2) * LDSW + ccol];
    float cz = sXYZ[(2 * 5 + 2) * LDSW + ccol];

    float w[25];
#pragma unroll
    for (int dy = 0; dy < 5; ++dy) {
        int yy = y - 2 + dy;
        bool yok = (yy >= 0) && (yy < HH);
#pragma unroll
        for (int dx = 0; dx < 5; ++dx) {
            int xx = x - 2 + dx;
            bool ok = yok && (xx >= 0) && (xx < WW);
            int col = t + 2 + dx;
            float ax = sXYZ[(0 * 5 + dy) * LDSW + col] - cx;
            float ay = sXYZ[(1 * 5 + dy) * LDSW + col] - cy;
            float az = sXYZ[(2 * 5 + dy) * LDSW + col] - cz;
            float d2 = ax * ax + ay * ay + az * az;
            float g  = __builtin_amdgcn_exp2f(-KEXP * d2);   // v_exp_f32
            float m  = (float)sMSK[dy * LDSW + col];         // 0.0 or 1.0
            w[dy * 5 + dx] = ok ? g * m : 0.0f;
        }
    }

    float* outBase = out + (size_t)(n * CC) * HH * WW + (size_t)y * WW + x;

    // ---------------- channel loop, double-buffered async softmax pipeline --------
    for (int c = 0; c < CC; ++c) {
        wait_async0();      // my channel-c slice has landed in LDS
        __syncthreads();    // everyone's slice landed; everyone done with buf[(c+1)&1]

        if (c + 1 < CC) {   // prefetch next channel while computing this one
            const float* sp = smBase + (size_t)(c + 1) * HH * WW;
            uint32_t dstL = ((c + 1) & 1) ? smL1 : smL0;
#pragma unroll
            for (int k = 0; k < 2; ++k)
                async_ld_b128(dstL + (uint32_t)(t + k * 256) * 16u, voff_plane[k], sp);
        }

        const float* buf = sSM[c & 1];
        float acc = 0.0f;
#pragma unroll
        for (int dy = 0; dy < 5; ++dy)
#pragma unroll
            for (int dx = 0; dx < 5; ++dx)
                acc += w[dy * 5 + dx] * buf[dy * LDSW + t + 2 + dx];

        outBase[(size_t)c * HH * WW] = acc;
    }
}

extern "C" void kernel_launch(void* const* d_in, const int* in_sizes, int n_in,
                              void* d_out, int out_size, void* d_ws, size_t ws_size,
                              hipStream_t stream) {
    const float* xyz         = (const float*)d_in[0];           // (4,3,64,2048) f32
    const float* sm          = (const float*)d_in[1];           // (4,20,64,2048) f32
    const unsigned char* msk = (const unsigned char*)d_in[2];   // (4,64,2048) bool
    float* out               = (float*)d_out;                   // (4,20,64,2048) f32

    dim3 grid(WW / TILE, HH, NN); // 8 x 64 x 4 = 2048 blocks
    bilateral5x5_kernel<<<grid, TILE, 0, stream>>>(xyz, sm, msk, out);
}